// HiPPOLTI_90658169684122
// MI455X (gfx1250) — compile-verified
//
#include <hip/hip_runtime.h>
#include <hip/hip_bf16.h>

#define CHANNELS 4096
#define HIDDEN   8192
#define MSPLIT   32              // row chunks for deterministic partial column sums
#define NGROUPS  (HIDDEN / 16)   // 512 groups of 16 columns
#define TILE     32

typedef __attribute__((ext_vector_type(2))) float v2f;
typedef __attribute__((ext_vector_type(8))) float v8f;

// ---------------------------------------------------------------------------
// Stage 1: partial column sums of A_d (HIDDEN x HIDDEN) using WMMA f32 16x16x4.
// colsum = ones(1xH) @ A_d  -> feed all-ones A operand; each WMMA accumulates
// 4 rows x 16 cols of A_d into the f32 accumulator (all D rows identical).
// One wave handles 16 columns over a 256-row chunk -> deterministic partials.
// ---------------------------------------------------------------------------
__global__ void hippo_colsum_partials(const float* __restrict__ A,
                                      float* __restrict__ part) {
    const int wave   = (blockIdx.x * blockDim.x + threadIdx.x) >> 5;
    const int lane   = threadIdx.x & 31;
    const int ngroup = wave & (NGROUPS - 1);   // which 16-column group
    const int mchunk = wave >> 9;              // which row chunk (0..MSPLIT-1)
    const int n      = ngroup * 16 + (lane & 15);
    const int rsel   = (lane >> 4) << 1;       // lanes 0-15: rows k0,k1 ; 16-31: k2,k3
    const int rows   = HIDDEN / MSPLIT;        // 256 rows per chunk
    const int mbase  = mchunk * rows;

    v8f acc = {0.f, 0.f, 0.f, 0.f, 0.f, 0.f, 0.f, 0.f};
    v2f a;
    a.x = 1.0f;  // all-ones A operand: D[i][j] = sum_k B[k][j] + C[i][j]
    a.y = 1.0f;

    for (int i = 0; i < rows; i += 4) {
        const size_t m = (size_t)(mbase + i + rsel);
        v2f b;
        b.x = A[m * HIDDEN + n];
        b.y = A[(m + 1) * HIDDEN + n];
        // 8 args: (neg_a, A, neg_b, B, c_mod, C, reuse_a, reuse_b)
        acc = __builtin_amdgcn_wmma_f32_16x16x4_f32(
            false, a, false, b, (short)0, acc, false, false);
    }

    // D layout: VGPR0, lanes 0-15 -> M=0, N=lane. All M rows equal -> colsum.
    if (lane < 16) {
        part[(size_t)mchunk * HIDDEN + n] = acc[0];
    }
}

// ---------------------------------------------------------------------------
// Stage 2: reduce MSPLIT partials -> colsum[HIDDEN] (deterministic order).
// ---------------------------------------------------------------------------
__global__ void hippo_colsum_reduce(const float* __restrict__ part,
                                    float* __restrict__ colsum) {
    const int n = blockIdx.x * blockDim.x + threadIdx.x;
    float s = 0.f;
    for (int k = 0; k < MSPLIT; ++k)
        s += part[(size_t)k * HIDDEN + n];
    colsum[n] = s;
}

// ---------------------------------------------------------------------------
// Stage 3: c_t[c,n] = colsum[n]*c_t_1[c,n] + B_d[n,c]*x[c]; write two copies.
// 32x32 tile; B_d transposed through padded LDS (stride 33 -> bank-conflict
// free); c_t_1 and out accessed as float4 along n (fully coalesced).
// ---------------------------------------------------------------------------
__global__ void hippo_fuse(const float* __restrict__ ct1,
                           const float* __restrict__ x,
                           const float* __restrict__ Bd,
                           const float* __restrict__ colsum,
                           float* __restrict__ out) {
    __shared__ float bt[TILE][TILE + 1];

    const int n0  = blockIdx.x * TILE;
    const int c0  = blockIdx.y * TILE;
    const int tid = threadIdx.x;  // 256 threads

    // Load B_d tile coalesced: rows are n (c fastest in memory).
    {
        const int col = tid & 31;   // c within tile
        const int row = tid >> 5;   // 0..7
        #pragma unroll
        for (int i = 0; i < 4; ++i) {
            const int r = row + i * 8;  // n within tile
            bt[r][col] = Bd[(size_t)(n0 + r) * CHANNELS + (c0 + col)];
        }
    }
    __syncthreads();

    const int tx = tid & 7;    // n quad index: n = n0 + tx*4
    const int ty = tid >> 3;   // c row: 0..31
    const int c  = c0 + ty;
    const int n  = n0 + tx * 4;

    const float  xv = x[c];
    const float4 cs = *reinterpret_cast<const float4*>(colsum + n);
    const size_t idx = (size_t)c * HIDDEN + n;
    const float4 ct  = *reinterpret_cast<const float4*>(ct1 + idx);

    float4 v;
    v.x = cs.x * ct.x + bt[tx * 4 + 0][ty] * xv;
    v.y = cs.y * ct.y + bt[tx * 4 + 1][ty] * xv;
    v.z = cs.z * ct.z + bt[tx * 4 + 2][ty] * xv;
    v.w = cs.w * ct.w + bt[tx * 4 + 3][ty] * xv;

    *reinterpret_cast<float4*>(out + idx) = v;
    *reinterpret_cast<float4*>(out + (size_t)CHANNELS * HIDDEN + idx) = v;
}

extern "C" void kernel_launch(void* const* d_in, const int* in_sizes, int n_in,
                              void* d_out, int out_size, void* d_ws, size_t ws_size,
                              hipStream_t stream) {
    // setup_inputs order: c_t_1 (C,H), x (C,), A_d (H,H), B_d (H,C) — all f32.
    const float* ct1 = (const float*)d_in[0];
    const float* x   = (const float*)d_in[1];
    const float* A_d = (const float*)d_in[2];
    const float* B_d = (const float*)d_in[3];
    float* out = (float*)d_out;

    // Scratch: partials [MSPLIT*HIDDEN] then colsum [HIDDEN]  (~1.03 MB).
    float* part   = (float*)d_ws;
    float* colsum = part + (size_t)MSPLIT * HIDDEN;

    // Stage 1: 512 col-groups * 32 row-chunks = 16384 waves; 8 waves/block.
    {
        const int total_waves = NGROUPS * MSPLIT;
        const int blocks = total_waves / 8;  // 2048 blocks of 256 threads
        hippo_colsum_partials<<<blocks, 256, 0, stream>>>(A_d, part);
    }
    // Stage 2: one thread per column.
    hippo_colsum_reduce<<<HIDDEN / 256, 256, 0, stream>>>(part, colsum);
    // Stage 3: 256 x 128 tiles of 32x32.
    {
        dim3 grid(HIDDEN / TILE, CHANNELS / TILE);
        hippo_fuse<<<grid, 256, 0, stream>>>(ct1, x, B_d, colsum, out);
    }
}